// OverlapAwareCircleLoss_56831007261038
// MI455X (gfx1250) — compile-verified
//
#include <hip/hip_runtime.h>
#include <hip/hip_bf16.h>
#include <math.h>

typedef __attribute__((ext_vector_type(2))) float v2f;
typedef __attribute__((ext_vector_type(8))) float v8f;

constexpr int B_ = 8, NS_ = 2048, NT_ = 2048, D_ = 256;
constexpr float POS_MARGIN = 0.1f, NEG_MARGIN = 1.4f;
constexpr float POS_OPTIMAL = 0.1f, NEG_OPTIMAL = 1.4f;
constexpr float LOG_SCALE = 24.0f, OVTH = 0.1f, NEGINF = -1.0e9f;
constexpr int LDT = 260;  // padded LDS row stride (floats); 260 % 64 == 4 -> conflict-free,
                          // 260 % 4 == 0 -> every row 16B-aligned for b128 LDS ops

__device__ __forceinline__ void lse_update(float& m, float& s, float v) {
  float M = fmaxf(m, v);
  s = s * __expf(m - M) + __expf(v - M);
  m = M;
}
__device__ __forceinline__ void lse_merge(float& m, float& s, float mo, float so) {
  float M = fmaxf(m, mo);
  s = s * __expf(m - M) + so * __expf(mo - M);
  m = M;
}

// ---------------------------------------------------------------------------
// Kernel 0: zero per-batch accumulators (must run every launch; graph-safe)
// ---------------------------------------------------------------------------
__global__ void oacl_init_kernel(float* rowTot, float* rowCnt, float* colTot, float* colCnt,
                                 unsigned* posAny, unsigned* negAny) {
  int i = threadIdx.x;
  if (i < B_) {
    rowTot[i] = 0.f; rowCnt[i] = 0.f; colTot[i] = 0.f; colCnt[i] = 0.f;
    posAny[i] = 0u; negAny[i] = 0u;
  }
}

// ---------------------------------------------------------------------------
// Kernel 1: per-row inverse L2 norms (wave-per-row, float4 loads, wave32)
// ---------------------------------------------------------------------------
__global__ void oacl_rownorm_kernel(const float* __restrict__ src, const float* __restrict__ tgt,
                                    float* __restrict__ invS, float* __restrict__ invT) {
  int wave = threadIdx.x >> 5, lane = threadIdx.x & 31;
  int row = blockIdx.x * 4 + wave;  // 0 .. B*NS + B*NT - 1
  const float* p; float* o;
  const int nSrc = B_ * NS_;
  if (row < nSrc) { p = src + (size_t)row * D_; o = invS + row; }
  else            { int r = row - nSrc; p = tgt + (size_t)r * D_; o = invT + r; }
  const float4* p4 = (const float4*)p;       // D_/4 = 64 float4 per row
  float4 a = p4[lane], b = p4[lane + 32];
  float ss = a.x*a.x + a.y*a.y + a.z*a.z + a.w*a.w
           + b.x*b.x + b.y*b.y + b.z*b.z + b.w*b.w;
  #pragma unroll
  for (int off = 16; off > 0; off >>= 1) ss += __shfl_xor(ss, off, 32);
  if (lane == 0) *o = 1.0f / fmaxf(sqrtf(ss), 1e-12f);
}

// ---------------------------------------------------------------------------
// Kernel 2: fused normalize-GEMM (fp32 WMMA) + circle-loss streaming logsumexp
// TRANS=false: rows=src, cols=tgt, ov[m][n]   (strideM=NT, strideN=1)
// TRANS=true : rows=tgt, cols=src, ov[n][m]   (contiguous float4 ov loads)
// ---------------------------------------------------------------------------
template <bool TRANS>
__global__ void __launch_bounds__(128)
oacl_circle_pass_kernel(const float* __restrict__ Afeat, const float* __restrict__ invA,
                        const float* __restrict__ Bfeat, const float* __restrict__ invB,
                        const float* __restrict__ ovl,
                        float* __restrict__ totAcc, float* __restrict__ cntAcc,
                        unsigned* __restrict__ posAnyG, unsigned* __restrict__ negAnyG) {
  constexpr int Mrows = TRANS ? NT_ : NS_;
  constexpr int Ncols = TRANS ? NS_ : NT_;

  __shared__ float As[16 * LDT];
  __shared__ float Bs[64 * LDT];
  __shared__ float mrg[4][16][4];
  __shared__ unsigned anyB[4][2][2];

  const int b = blockIdx.y;
  const int m0 = blockIdx.x * 16;
  const int tid = threadIdx.x;
  const int w = tid >> 5, lane = tid & 31;
  const int half = lane >> 4, lm = lane & 15;

  const float* Ab = Afeat + (size_t)b * Mrows * D_;
  const float* Bb = Bfeat + (size_t)b * Ncols * D_;
  const float* ovb = ovl + (size_t)b * NS_ * NT_;
  const float* iA = invA + (size_t)b * Mrows;
  const float* iB = invB + (size_t)b * Ncols;

  // Stage normalized A tile (16 x D) into LDS via float4 (b128) ops.
  for (int idx4 = tid; idx4 < 16 * (D_ / 4); idx4 += 128) {
    int r = idx4 >> 6, c4 = idx4 & 63;
    float4 v = ((const float4*)(Ab + (size_t)(m0 + r) * D_))[c4];
    float s = iA[m0 + r];
    v.x *= s; v.y *= s; v.z *= s; v.w *= s;
    *(float4*)&As[r * LDT + c4 * 4] = v;
  }

  // Streaming logsumexp state: slot j <-> tile row (j + 8*half) per WMMA C layout.
  float mp[8], sp[8], mn[8], sn[8];
  #pragma unroll
  for (int j = 0; j < 8; ++j) { mp[j] = -INFINITY; sp[j] = 0.f; mn[j] = -INFINITY; sn[j] = 0.f; }
  unsigned pbits = 0u, nbits = 0u;

  for (int nt0 = 0; nt0 < Ncols; nt0 += 64) {
    __syncthreads();
    // Stage normalized B tile (64 x D) into LDS via float4 (b128) ops.
    for (int idx4 = tid; idx4 < 64 * (D_ / 4); idx4 += 128) {
      int r = idx4 >> 6, c4 = idx4 & 63;
      float4 v = ((const float4*)(Bb + (size_t)(nt0 + r) * D_))[c4];
      float s = iB[nt0 + r];
      v.x *= s; v.y *= s; v.z *= s; v.w *= s;
      *(float4*)&Bs[r * LDT + c4 * 4] = v;
    }
    // Prefetch next B tile into caches (global_prefetch_b8).
    {
      int pr = nt0 + 64 + (tid >> 1);
      if (pr < Ncols) __builtin_prefetch(&Bb[(size_t)pr * D_ + (tid & 1) * 128], 0, 0);
    }
    __syncthreads();

    const int n = nt0 + w * 16 + lm;   // this lane's output column
    v8f acc = {0.f, 0.f, 0.f, 0.f, 0.f, 0.f, 0.f, 0.f};
    const float* aRow = &As[lm * LDT + 2 * half];
    const float* bRow = &Bs[(w * 16 + lm) * LDT + 2 * half];
    // A frag: A[m=lm][k = kb + 2*half + v]; B frag: B[k = kb + 2*half + v][n=lm]
    #pragma unroll 8
    for (int k = 0; k < D_; k += 4) {
      v2f a  = *(const v2f*)(aRow + k);
      v2f bb = *(const v2f*)(bRow + k);
      acc = __builtin_amdgcn_wmma_f32_16x16x4_f32(false, a, false, bb, (short)0, acc,
                                                  false, false);
    }

    // Load this lane's 8 overlap values (compile-time strides -> shifts).
    float ovv[8];
    if constexpr (TRANS) {
      // ov[n][m0 + 8*half + j], contiguous in j -> two b128 loads
      const float4* p = (const float4*)(ovb + (size_t)n * NT_ + (m0 + 8 * half));
      float4 o0 = p[0], o1 = p[1];
      ovv[0] = o0.x; ovv[1] = o0.y; ovv[2] = o0.z; ovv[3] = o0.w;
      ovv[4] = o1.x; ovv[5] = o1.y; ovv[6] = o1.z; ovv[7] = o1.w;
    } else {
      #pragma unroll
      for (int j = 0; j < 8; ++j)
        ovv[j] = ovb[(size_t)(m0 + j + 8 * half) * NT_ + n];
    }

    // Epilogue: sim -> dist -> logits -> streaming logsumexp.
    #pragma unroll
    for (int j = 0; j < 8; ++j) {
      float ov = ovv[j];
      float sim = acc[j];
      float d = sqrtf(fmaxf(2.f - 2.f * sim, 1e-12f));
      bool pos = ov >= OVTH;
      float ps = sqrtf(fmaxf(ov, 1e-8f));
      float pw = fmaxf(d - POS_OPTIMAL, 0.f) * ps;           // detached weights
      float nw = fmaxf(NEG_OPTIMAL - d, 0.f);
      float lp = LOG_SCALE * (d - POS_MARGIN) * pw + (pos ? 0.f : NEGINF);
      float ln = LOG_SCALE * (NEG_MARGIN - d) * nw + (pos ? NEGINF : 0.f);
      lse_update(mp[j], sp[j], lp);
      lse_update(mn[j], sn[j], ln);
      pbits |= (pos ? 1u : 0u) << j;
      nbits |= (pos ? 0u : 1u) << j;
    }
  }

  // Merge the 16 column-lanes of each half (same rows, disjoint columns).
  #pragma unroll
  for (int j = 0; j < 8; ++j) {
    #pragma unroll
    for (int off = 1; off < 16; off <<= 1) {
      float mo = __shfl_xor(mp[j], off, 32), so = __shfl_xor(sp[j], off, 32);
      lse_merge(mp[j], sp[j], mo, so);
      mo = __shfl_xor(mn[j], off, 32); so = __shfl_xor(sn[j], off, 32);
      lse_merge(mn[j], sn[j], mo, so);
    }
  }
  #pragma unroll
  for (int off = 1; off < 16; off <<= 1) {
    pbits |= __shfl_xor(pbits, off, 32);
    nbits |= __shfl_xor(nbits, off, 32);
  }
  if (lm == 0) {
    #pragma unroll
    for (int j = 0; j < 8; ++j) {
      int r = j + 8 * half;
      mrg[w][r][0] = mp[j]; mrg[w][r][1] = sp[j];
      mrg[w][r][2] = mn[j]; mrg[w][r][3] = sn[j];
    }
    anyB[w][half][0] = pbits; anyB[w][half][1] = nbits;
  }
  __syncthreads();

  // Wave 0, lanes 0..15: merge the 4 waves' disjoint-column states per row.
  if (w == 0 && lane < 16) {
    int r = lane, h = r >> 3, bit = r & 7;
    float Mp = mrg[0][r][0], Sp = mrg[0][r][1], Mn = mrg[0][r][2], Sn = mrg[0][r][3];
    unsigned pa = (anyB[0][h][0] >> bit) & 1u, na = (anyB[0][h][1] >> bit) & 1u;
    #pragma unroll
    for (int ww = 1; ww < 4; ++ww) {
      lse_merge(Mp, Sp, mrg[ww][r][0], mrg[ww][r][1]);
      lse_merge(Mn, Sn, mrg[ww][r][2], mrg[ww][r][3]);
      pa |= (anyB[ww][h][0] >> bit) & 1u;
      na |= (anyB[ww][h][1] >> bit) & 1u;
    }
    float lp = Mp + __logf(Sp);
    float ln = Mn + __logf(Sn);
    float x = lp + ln;
    float spx = (x > 20.f) ? x : log1pf(__expf(x));   // softplus
    float loss = spx / LOG_SCALE;
    if (pa && na) {
      atomicAdd(&totAcc[b], loss);
      atomicAdd(&cntAcc[b], 1.0f);
    }
    if (pa) atomicOr(&posAnyG[b], 1u);
    if (na) atomicOr(&negAnyG[b], 1u);
  }
}

// ---------------------------------------------------------------------------
// Kernel 3: final per-batch means + validity + global mean -> scalar
// ---------------------------------------------------------------------------
__global__ void oacl_finalize_kernel(const float* rowTot, const float* rowCnt,
                                     const float* colTot, const float* colCnt,
                                     const unsigned* posAny, const unsigned* negAny,
                                     float* out) {
  if (threadIdx.x != 0 || blockIdx.x != 0) return;
  float sum = 0.f; float nValid = 0.f;
  for (int b = 0; b < B_; ++b) {
    float rl = (rowCnt[b] > 0.f) ? rowTot[b] / fmaxf(rowCnt[b], 1.f) : 0.f;
    float cl = (colCnt[b] > 0.f) ? colTot[b] / fmaxf(colCnt[b], 1.f) : 0.f;
    float bl = 0.5f * (rl + cl);
    bool finite = (bl == bl) && (fabsf(bl) < 3.0e38f);
    bool valid = (posAny[b] != 0u) && (negAny[b] != 0u) && finite;
    if (valid) { sum += bl; nValid += 1.f; }
  }
  out[0] = sum / fmaxf(nValid, 1.f);
}

// ---------------------------------------------------------------------------
extern "C" void kernel_launch(void* const* d_in, const int* in_sizes, int n_in,
                              void* d_out, int out_size, void* d_ws, size_t ws_size,
                              hipStream_t stream) {
  const float* src = (const float*)d_in[0];  // (B, NS, D)
  const float* tgt = (const float*)d_in[1];  // (B, NT, D)
  const float* ovl = (const float*)d_in[2];  // (B, NS, NT)
  float* out = (float*)d_out;

  float* invS   = (float*)d_ws;                    // B*NS
  float* invT   = invS + (size_t)B_ * NS_;         // B*NT
  float* rowTot = invT + (size_t)B_ * NT_;         // B
  float* rowCnt = rowTot + B_;
  float* colTot = rowCnt + B_;
  float* colCnt = colTot + B_;
  unsigned* posAny = (unsigned*)(colCnt + B_);
  unsigned* negAny = posAny + B_;

  hipLaunchKernelGGL(oacl_init_kernel, dim3(1), dim3(64), 0, stream,
                     rowTot, rowCnt, colTot, colCnt, posAny, negAny);

  const int totalRows = B_ * (NS_ + NT_);
  hipLaunchKernelGGL(oacl_rownorm_kernel, dim3(totalRows / 4), dim3(128), 0, stream,
                     src, tgt, invS, invT);

  // Row-direction pass: rows = src, cols = tgt.
  hipLaunchKernelGGL((oacl_circle_pass_kernel<false>), dim3(NS_ / 16, B_), dim3(128), 0, stream,
                     src, invS, tgt, invT, ovl, rowTot, rowCnt, posAny, negAny);
  // Col-direction pass: rows = tgt, cols = src (transposed overlaps access).
  hipLaunchKernelGGL((oacl_circle_pass_kernel<true>), dim3(NT_ / 16, B_), dim3(128), 0, stream,
                     tgt, invT, src, invS, ovl, colTot, colCnt, posAny, negAny);

  hipLaunchKernelGGL(oacl_finalize_kernel, dim3(1), dim3(1), 0, stream,
                     rowTot, rowCnt, colTot, colCnt, posAny, negAny, out);
}